// GCNLayer_26431228740344
// MI455X (gfx1250) — compile-verified
//
#include <hip/hip_runtime.h>
#include <hip/hip_bf16.h>

typedef __attribute__((ext_vector_type(2))) float v2f;
typedef __attribute__((ext_vector_type(8))) float v8f;

#define D 128  // D_IN == D_OUT == 128

// Guaranteed hardware fp32 atomic add (no-return form, STOREcnt-tracked,
// executed in the L2 atomic units). Avoids any risk of a CAS-loop lowering.
__device__ __forceinline__ void atomic_add_f32_hw(float* addr, float val) {
    asm volatile("global_atomic_add_f32 %0, %1, off"
                 :
                 : "v"(addr), "v"(val)
                 : "memory");
}

// ---------------------------------------------------------------------------
// Kernel 0: zero the accumulator (d_out) — harness poisons it with 0xAA.
// ---------------------------------------------------------------------------
__global__ __launch_bounds__(256) void gcn_zero(float4* __restrict__ p, int n4) {
    int i = blockIdx.x * blockDim.x + threadIdx.x;
    if (i < n4) p[i] = make_float4(0.f, 0.f, 0.f, 0.f);
}

// ---------------------------------------------------------------------------
// Kernel 1: support = x @ W  via V_WMMA_F32_16X16X4_F32 (fp32 matrix pipe).
// Block = 256 threads = 8 waves. Block owns a 16-row stripe; wave w owns the
// 16-column tile n0 = 16*w, so one block covers 16 x 128 of the output.
// K loop: 128/4 = 32 WMMA ops per tile.
//
// A (16x4 f32) per ISA: lanes 0-15 -> M=lane, {K, K+1}; lanes 16-31 -> {K+2, K+3}.
// B (4x16 f32): symmetric — lanes 0-15 -> N=lane, rows {K, K+1}; lanes 16-31 -> {K+2, K+3}.
// C/D (16x16 f32): VGPR v, lanes 0-15 -> M=v, N=lane; lanes 16-31 -> M=v+8.
// ---------------------------------------------------------------------------
__global__ __launch_bounds__(256) void gcn_gemm_wmma(const float* __restrict__ x,
                                                     const float* __restrict__ W,
                                                     float* __restrict__ support) {
    const int lane = threadIdx.x & 31;
    const int wave = threadIdx.x >> 5;     // 0..7 -> N tile
    const int m0   = blockIdx.x * 16;      // 6250 blocks cover 100000 rows exactly
    const int n0   = wave * 16;
    const int half = lane >> 4;            // 0: K,K+1   1: K+2,K+3
    const int l16  = lane & 15;

    const float* arow = x + (size_t)(m0 + l16) * D;  // this lane's A row
    v8f c = {};
#pragma unroll 4
    for (int k = 0; k < D; k += 4) {
        const int ka = k + half * 2;
        v2f a;
        a.x = arow[ka];
        a.y = arow[ka + 1];
        v2f b;
        b.x = W[(size_t)ka * D + n0 + l16];
        b.y = W[(size_t)(ka + 1) * D + n0 + l16];
        // 8 args: (neg_a, A, neg_b, B, c_mod, C, reuse_a, reuse_b)
        c = __builtin_amdgcn_wmma_f32_16x16x4_f32(
                false, a, false, b, (short)0, c, false, false);
    }

    float* srow = support + n0 + l16;
#pragma unroll
    for (int v = 0; v < 8; ++v) {
        srow[(size_t)(m0 + v + half * 8) * D] = c[v];
    }
}

// ---------------------------------------------------------------------------
// Kernel 2: COO scatter-add. One wave per edge. The edge index is derived from
// blockIdx.x and a readfirstlane'd wave id, so (r, c, v) are provably wave-
// uniform -> scalar s_load of the edge descriptors. Each lane owns one float4
// chunk of the 128-wide row: global_load_b128 gather (512 B coalesced per edge
// from L2-resident support) + 4 no-return global_atomic_add_f32.
// ---------------------------------------------------------------------------
__global__ __launch_bounds__(256) void gcn_scatter_atomic(const float* __restrict__ support,
                                                          const int* __restrict__ erow,
                                                          const int* __restrict__ ecol,
                                                          const float* __restrict__ eval,
                                                          float* __restrict__ out,
                                                          int nEdges) {
    const int wv = __builtin_amdgcn_readfirstlane(threadIdx.x >> 5);  // 0..7, wave-uniform
    const int e  = blockIdx.x * 8 + wv;
    if (e >= nEdges) return;
    const int lane = threadIdx.x & 31;

    const int   r = erow[e];   // wave-uniform -> SMEM
    const int   c = ecol[e];
    const float v = eval[e];

    const float4 m = *((const float4*)(support + (size_t)c * D) + lane);
    float* dst = out + (size_t)r * D + lane * 4;

    atomic_add_f32_hw(dst + 0, m.x * v);
    atomic_add_f32_hw(dst + 1, m.y * v);
    atomic_add_f32_hw(dst + 2, m.z * v);
    atomic_add_f32_hw(dst + 3, m.w * v);
}

// ---------------------------------------------------------------------------
// Kernel 3: out = tanh(out), in place, float4-vectorized.
// ---------------------------------------------------------------------------
__global__ __launch_bounds__(256) void gcn_tanh(float4* __restrict__ p, int n4) {
    int i = blockIdx.x * blockDim.x + threadIdx.x;
    if (i < n4) {
        float4 t = p[i];
        t.x = tanhf(t.x);
        t.y = tanhf(t.y);
        t.z = tanhf(t.z);
        t.w = tanhf(t.w);
        p[i] = t;
    }
}

extern "C" void kernel_launch(void* const* d_in, const int* in_sizes, int n_in,
                              void* d_out, int out_size, void* d_ws, size_t ws_size,
                              hipStream_t stream) {
    const float* x    = (const float*)d_in[0];   // [N, 128]
    const float* W    = (const float*)d_in[1];   // [128, 128]
    const int*   erow = (const int*)d_in[2];     // [E]
    const int*   ecol = (const int*)d_in[3];     // [E]
    const float* eval = (const float*)d_in[4];   // [E]
    float* out = (float*)d_out;                  // [N, 128]

    const int nNodes = in_sizes[0] / D;          // 100000
    const int nEdges = in_sizes[2];              // 1600000

    float* support = (float*)d_ws;               // [N, 128] scratch (51.2 MB)

    // 1) out = 0 (accumulator; also erases harness poison)
    {
        const int n4 = out_size / 4;
        gcn_zero<<<(n4 + 255) / 256, 256, 0, stream>>>((float4*)out, n4);
    }
    // 2) support = x @ W   (fp32 WMMA)
    {
        const int blocks = nNodes / 16;          // 100000/16 = 6250, exact
        gcn_gemm_wmma<<<blocks, 256, 0, stream>>>(x, W, support);
    }
    // 3) out[r] += v * support[c]   (one wave per edge, hw fp32 L2 atomics)
    {
        const int blocks = (nEdges + 7) / 8;     // 8 edges (waves) per block
        gcn_scatter_atomic<<<blocks, 256, 0, stream>>>(support, erow, ecol, eval, out, nEdges);
    }
    // 4) out = tanh(out)
    {
        const int n4 = out_size / 4;
        gcn_tanh<<<(n4 + 255) / 256, 256, 0, stream>>>((float4*)out, n4);
    }
}